// BCIM_45861660787130
// MI455X (gfx1250) — compile-verified
//
#include <hip/hip_runtime.h>
#include <cstdint>

typedef float v2f __attribute__((ext_vector_type(2)));
typedef float v8f __attribute__((ext_vector_type(8)));

#define NCH        128
#define NPIX       1024        // 32*32 pixels per image
#define PRODSTRIDE 144         // pad so half-waves hit disjoint LDS banks
#define TILESTRIDE 36          // 32 pixels + pad, keeps 16B alignment for b128 ops

// Low 32 bits of a generic pointer to __shared__ = LDS byte offset (aperture mapping).
__device__ __forceinline__ uint32_t lds_off(const void* p) {
  return (uint32_t)(uintptr_t)p;
}

// 16B per lane, global -> LDS, tracked by ASYNCcnt (CDNA5 async data path).
__device__ __forceinline__ void async_copy16(uint32_t lds_byte, const float* g) {
  asm volatile("global_load_async_to_lds_b128 %0, %1, off"
               :: "v"(lds_byte), "v"((uint64_t)(uintptr_t)g)
               : "memory");
}
__device__ __forceinline__ void wait_async_le1() {
  asm volatile("s_wait_asynccnt 1" ::: "memory");
}
__device__ __forceinline__ void wait_async_le0() {
  asm volatile("s_wait_asynccnt 0" ::: "memory");
}

__global__ __launch_bounds__(256) void bcim_kernel(const float* __restrict__ src_all,
                                                   float* __restrict__ out_all) {
  __shared__ float rn[NPIX];                   // 1/||p||  per pixel
  __shared__ float plane[2][NPIX];             // double-buffered channel plane (async dst)
  __shared__ float rowb[2][NPIX];              // row box sums (double buffered)
  __shared__ float prodb[8][4 * PRODSTRIDE];   // per-wave unit*boxsum products (4 channels)
  __shared__ float simb[NPIX];                 // final similarity per pixel
  __shared__ float tile[NCH * TILESTRIDE];     // pass-3 transpose tile (32 pixels x 128 ch)

  const int t    = threadIdx.x;
  const int lane = t & 31;
  const int w    = t >> 5;               // wave id, 8 waves
  const int b    = blockIdx.x;
  const float* src = src_all + (size_t)b * (NCH * NPIX);
  const int p0   = 4 * t;                // this thread's 4 pixels p0..p0+3 (one image row chunk)
  const int col  = p0 & 31;              // column of first pixel (multiple of 4)
  const int h    = p0 >> 5;              // image row (group never crosses rows)

  // Kick off async staging of channel 0 plane; overlaps with the norm pass.
  async_copy16(lds_off(&plane[0][p0]), src + p0);

  // ---------------- pass 1: per-pixel L2 norm over 128 channels ----------------
  float nx = 0.f, ny = 0.f, nz = 0.f, nw_ = 0.f;
  for (int c = 0; c < NCH; ++c) {
    const float4 v = *(const float4*)(src + c * NPIX + p0);
    nx += v.x * v.x; ny += v.y * v.y; nz += v.z * v.z; nw_ += v.w * v.w;
  }
  float4 rn4;
  rn4.x = 1.0f / sqrtf(nx);
  rn4.y = 1.0f / sqrtf(ny);
  rn4.z = 1.0f / sqrtf(nz);
  rn4.w = 1.0f / sqrtf(nw_);
  *(float4*)&rn[p0] = rn4;   // neighbors that read this are in the same wave (in-order LDS)

  // ---------------- pass 2: sim = <unit, 3x3 boxsum(unit)> / 9 ----------------
  v8f acc[8];
  const v8f vzero = {0.f, 0.f, 0.f, 0.f, 0.f, 0.f, 0.f, 0.f};
#pragma unroll
  for (int g = 0; g < 8; ++g) acc[g] = vzero;
  const v2f bones = {1.0f, 1.0f};        // B = ones -> WMMA row-sum reduction
  const int m  = lane & 15;              // pixel row within 16-pixel WMMA group
  const int kb = (lane >> 4) * 2;        // A-matrix K base per ISA lane layout
#if !__has_builtin(__builtin_amdgcn_wmma_f32_16x16x4_f32)
  float4 simacc = make_float4(0.f, 0.f, 0.f, 0.f);
#endif

  for (int r = 0; r < 32; ++r) {
#pragma unroll
    for (int cc = 0; cc < 4; ++cc) {
      const int c   = 4 * r + cc;
      const int cur = cc & 1;            // == c&1 (4r even) -> compile-time buffer select
      if (c + 1 < NCH) {
        async_copy16(lds_off(&plane[cur ^ 1][p0]), src + (c + 1) * NPIX + p0);
        wait_async_le1();                // channel c slice complete (wave-local prod/cons)
      } else {
        wait_async_le0();
      }
      // row sums (all reads wave-local: wave owns 4 complete image rows)
      const float4 rv  = *(const float4*)&plane[cur][p0];
      const float4 rnv = *(const float4*)&rn[p0];
      const float ux = rv.x * rnv.x, uy = rv.y * rnv.y;
      const float uz = rv.z * rnv.z, uw = rv.w * rnv.w;
      float ul = 0.f, ur_ = 0.f;
      if (col > 0)  ul  = plane[cur][p0 - 1] * rn[p0 - 1];
      if (col < 28) ur_ = plane[cur][p0 + 4] * rn[p0 + 4];
      float4 rs;
      rs.x = ul + ux + uy;
      rs.y = ux + uy + uz;
      rs.z = uy + uz + uw;
      rs.w = uz + uw + ur_;
      *(float4*)&rowb[cur][p0] = rs;
      __syncthreads();                   // rowb[cur] complete across waves (also fences reuse)
      // column sums (cross-wave reads of rowb)
      float sx = rs.x, sy = rs.y, sz = rs.z, sw = rs.w;
      if (h > 0) {
        const float4 a = *(const float4*)&rowb[cur][p0 - 32];
        sx += a.x; sy += a.y; sz += a.z; sw += a.w;
      }
      if (h < 31) {
        const float4 a = *(const float4*)&rowb[cur][p0 + 32];
        sx += a.x; sy += a.y; sz += a.z; sw += a.w;
      }
      float4 pr;
      pr.x = ux * sx; pr.y = uy * sy; pr.z = uz * sz; pr.w = uw * sw;
      *(float4*)&prodb[w][cc * PRODSTRIDE + 4 * lane] = pr;   // wave-local buffer
    }
    // WMMA ones-reduction: D[m,n] += sum_k A[m,k]; 16 pixels x 4 channels per call, f32 exact.
#if __has_builtin(__builtin_amdgcn_wmma_f32_16x16x4_f32)
#pragma unroll
    for (int g = 0; g < 8; ++g) {
      v2f a;
      a.x = prodb[w][kb       * PRODSTRIDE + 16 * g + m];   // A[m, K=kb]
      a.y = prodb[w][(kb + 1) * PRODSTRIDE + 16 * g + m];   // A[m, K=kb+1]
      acc[g] = __builtin_amdgcn_wmma_f32_16x16x4_f32(
          false, a, false, bones, (short)0, acc[g], false, false);
    }
#else
#pragma unroll
    for (int cc = 0; cc < 4; ++cc) {
      simacc.x += prodb[w][cc * PRODSTRIDE + 4 * lane + 0];
      simacc.y += prodb[w][cc * PRODSTRIDE + 4 * lane + 1];
      simacc.z += prodb[w][cc * PRODSTRIDE + 4 * lane + 2];
      simacc.w += prodb[w][cc * PRODSTRIDE + 4 * lane + 3];
    }
#endif
  }

  // sim extraction. C/D layout: lane 0 v[r] = D[M=r,N=0]; lane 16 v[r] = D[M=8+r,N=0].
  const float inv9 = 1.0f / 9.0f;
#if __has_builtin(__builtin_amdgcn_wmma_f32_16x16x4_f32)
  if (lane == 0 || lane == 16) {
    const int mb = (lane >> 4) * 8;
#pragma unroll
    for (int g = 0; g < 8; ++g) {
#pragma unroll
      for (int q = 0; q < 8; ++q) {
        simb[128 * w + 16 * g + mb + q] = acc[g][q] * inv9;
      }
    }
  }
#else
  simb[p0 + 0] = simacc.x * inv9;
  simb[p0 + 1] = simacc.y * inv9;
  simb[p0 + 2] = simacc.z * inv9;
  simb[p0 + 3] = simacc.w * inv9;
#endif

  // ---------------- pass 3: out[b, j*1024+p, k] = src[b, 2k+j, p] * sim[p] ----------------
  // LDS transpose tiles: 32 pixels x 128 channels; async global->LDS, coalesced f4 stores out.
  const size_t outbase = (size_t)b * (2048ull * 64ull);
  const int rr   = t >> 2;        // output row within tile: 0..63
  const int part = t & 3;         // quarter of the 64-float row
  const int j    = rr >> 5;       // interleave index (c = 2k + j)
  const int pl   = rr & 31;       // pixel within tile
  for (int tb = 0; tb < 32; ++tb) {
    const int pb = 32 * tb;
#pragma unroll
    for (int rep = 0; rep < 4; ++rep) {
      const int c  = 32 * rep + (t >> 3);
      const int px = 4 * (t & 7);
      async_copy16(lds_off(&tile[c * TILESTRIDE + px]), src + c * NPIX + pb + px);
    }
    wait_async_le0();
    __syncthreads();              // tile fully staged by all waves
    const float s = simb[pb + pl];
#pragma unroll
    for (int i = 0; i < 4; ++i) {
      const int k0 = 4 * part + 16 * i;   // 4 lanes/row cover 64B contiguous per store
      float4 o;
      o.x = tile[(j + 2 * (k0 + 0)) * TILESTRIDE + pl] * s;
      o.y = tile[(j + 2 * (k0 + 1)) * TILESTRIDE + pl] * s;
      o.z = tile[(j + 2 * (k0 + 2)) * TILESTRIDE + pl] * s;
      o.w = tile[(j + 2 * (k0 + 3)) * TILESTRIDE + pl] * s;
      *(float4*)(out_all + outbase + ((size_t)(j * NPIX + pb + pl)) * 64 + k0) = o;
    }
    __syncthreads();              // protect tile reuse next iteration
  }
}

extern "C" void kernel_launch(void* const* d_in, const int* in_sizes, int n_in,
                              void* d_out, int out_size, void* d_ws, size_t ws_size,
                              hipStream_t stream) {
  const float* p = (const float*)d_in[0];
  float* out = (float*)d_out;
  const int B = in_sizes[0] / (NCH * NPIX);   // 256 for the reference shapes
  bcim_kernel<<<B, 256, 0, stream>>>(p, out);
  (void)n_in; (void)out_size; (void)d_ws; (void)ws_size;
}